// SeesawLoss_prior_32418413150371
// MI455X (gfx1250) — compile-verified
//
#include <hip/hip_runtime.h>
#include <stdint.h>

#define TPB 256

// Low 32 bits of a flat pointer to a __shared__ object are the LDS byte offset
// (the flat LDS aperture base sits in the high 32 bits).
__device__ __forceinline__ uint32_t lds_off_of(const void* p) {
    return (uint32_t)(uintptr_t)p;
}

// CDNA5 async global->LDS copy, 8 bytes per lane (rows are 8B-aligned, not 16B).
// VDST = per-lane LDS byte address, VADDR = per-lane 64-bit global address.
// Ordering vs. the consuming LDS reads is enforced by the volatile asm +
// "memory" clobber here and on the s_wait_asynccnt asm.
__device__ __forceinline__ void async_ld_b64(uint32_t ldsb, const void* g) {
    asm volatile("global_load_async_to_lds_b64 %0, %1, off"
                 :
                 : "v"(ldsb), "v"(g)
                 : "memory");
}

__device__ __forceinline__ void merge_ms(float& m, float& s, float mo, float so) {
    float nm = fmaxf(m, mo);
    s = s * __expf(m - nm) + so * __expf(mo - nm);
    m = nm;
}

__global__ __launch_bounds__(TPB)
void seesaw_rows(const float* __restrict__ x, const int* __restrict__ tgt,
                 const float* __restrict__ w, float* __restrict__ rowres, int C)
{
    __shared__ __align__(16) float shx[2][TPB * 2];
    __shared__ __align__(16) float shw[2][TPB * 2];
    __shared__ float rm[TPB / 32], rs[TPB / 32];

    const int b   = blockIdx.x;
    const int tid = threadIdx.x;
    const int t   = tgt[b];
    const float* __restrict__ px = x + (size_t)b * (size_t)C;
    const float* __restrict__ pw = w + (size_t)t * (size_t)C;

    const int n2 = C >> 1;                   // # of float2 chunks (C is even here)
    const int nt = (n2 + TPB - 1) / TPB;     // # of tiles

    float m = -3.402823466e38f;              // -FLT_MAX
    float s = 0.0f;

    // ---- issue tile 0 (clamped addresses keep EXEC full -> uniform ASYNCcnt) ----
    {
        int idx = tid;
        int c2  = idx < n2 ? idx : (n2 - 1);
        async_ld_b64(lds_off_of(&shx[0][tid * 2]), (const void*)(px + (size_t)c2 * 2));
        async_ld_b64(lds_off_of(&shw[0][tid * 2]), (const void*)(pw + (size_t)c2 * 2));
    }

    for (int i = 0; i < nt; ++i) {
        if (i + 1 < nt) {
            // issue next tile into the other buffer, then wait for current tile
            int buf = (i + 1) & 1;
            int idx = (i + 1) * TPB + tid;
            int c2  = idx < n2 ? idx : (n2 - 1);
            async_ld_b64(lds_off_of(&shx[buf][tid * 2]), (const void*)(px + (size_t)c2 * 2));
            async_ld_b64(lds_off_of(&shw[buf][tid * 2]), (const void*)(pw + (size_t)c2 * 2));
            // 4 outstanding; async loads complete in order => <=2 left means tile i landed
            asm volatile("s_wait_asynccnt 2" ::: "memory");
        } else {
            asm volatile("s_wait_asynccnt 0" ::: "memory");
        }

        int buf = i & 1;
        float2 xv = *(const float2*)&shx[buf][tid * 2];
        float2 wv = *(const float2*)&shw[buf][tid * 2];
        int idx = i * TPB + tid;
        if (idx < n2) {
            // online softmax with exp(x+logw) = w*exp(x); W<=1 keeps this stable
            float nm = fmaxf(m, fmaxf(xv.x, xv.y));
            s = s * __expf(m - nm)
              + wv.x * __expf(xv.x - nm)
              + wv.y * __expf(xv.y - nm);
            m = nm;
        }
    }

    if ((C & 1) && tid == 0) {               // generic odd-C tail (unused for C=8142)
        float xv = px[C - 1], wv = pw[C - 1];
        float nm = fmaxf(m, xv);
        s = s * __expf(m - nm) + wv * __expf(xv - nm);
        m = nm;
    }

    // ---- wave32 reduction (xor tree), then cross-wave merge ----
    #pragma unroll
    for (int off = 16; off >= 1; off >>= 1) {
        float mo = __shfl_xor(m, off, 32);
        float so = __shfl_xor(s, off, 32);
        merge_ms(m, s, mo, so);
    }
    int wv_id = tid >> 5;
    if ((tid & 31) == 0) { rm[wv_id] = m; rs[wv_id] = s; }
    __syncthreads();
    if (tid == 0) {
        #pragma unroll
        for (int k = 1; k < TPB / 32; ++k) merge_ms(m, s, rm[k], rs[k]);
        float lse = m + logf(s);
        // target logit = x[b,t] + log W[t,t]  (W[t,t]==1, but compute exactly)
        rowres[b] = lse - (px[t] + logf(pw[t]));
    }
}

__global__ __launch_bounds__(TPB)
void seesaw_finalize(const float* __restrict__ rowres, float* __restrict__ out, int B)
{
    __shared__ float ps[TPB / 32];
    float sum = 0.0f;
    for (int i = threadIdx.x; i < B; i += TPB) sum += rowres[i];
    #pragma unroll
    for (int off = 16; off >= 1; off >>= 1) sum += __shfl_xor(sum, off, 32);
    int wv_id = threadIdx.x >> 5;
    if ((threadIdx.x & 31) == 0) ps[wv_id] = sum;
    __syncthreads();
    if (threadIdx.x == 0) {
        float tot = 0.0f;
        #pragma unroll
        for (int k = 0; k < TPB / 32; ++k) tot += ps[k];
        out[0] = tot / (float)B;
    }
}

extern "C" void kernel_launch(void* const* d_in, const int* in_sizes, int n_in,
                              void* d_out, int out_size, void* d_ws, size_t ws_size,
                              hipStream_t stream) {
    const float* x      = (const float*)d_in[0];   // [B, C] fp32
    const int*   target = (const int*)d_in[1];     // [B]    int32
    const float* wmat   = (const float*)d_in[2];   // [C, C] fp32

    const int B = in_sizes[1];
    const int C = in_sizes[0] / B;

    float* rowres = (float*)d_ws;                  // B floats of scratch

    seesaw_rows<<<B, TPB, 0, stream>>>(x, target, wmat, rowres, C);
    seesaw_finalize<<<1, TPB, 0, stream>>>(rowres, (float*)d_out, B);
}